// GATEncoder_81449759801844
// MI455X (gfx1250) — compile-verified
//
#include <hip/hip_runtime.h>

// ---------------------------------------------------------------------------
// GAT encoder (3 layers, H=4 heads, D=128) for MI455X / gfx1250.
//   per layer:  h = x@W (WMMA f32)  -> att logits -> edge softmax (3 passes,
//   atomic max/add in L2) -> bias+BN+ELU -> next layer; final mean pool.
// ---------------------------------------------------------------------------

#define Dch      128
#define Hh       4
#define Ch       32          // Dch / Hh
#define Lnum     3
#define NEG_SLOPE 0.2f
#define BN_EPS   1e-5f

typedef __attribute__((ext_vector_type(2))) float v2f;
typedef __attribute__((ext_vector_type(8))) float v8f;

// ---- monotonic float<->uint key for atomicMax on floats -------------------
__device__ __forceinline__ unsigned fkey(float f) {
  unsigned b = __float_as_uint(f);
  return b ^ ((unsigned)((int)b >> 31) | 0x80000000u);
}
__device__ __forceinline__ float kinv(unsigned k) {
  return __uint_as_float((k & 0x80000000u) ? (k ^ 0x80000000u) : ~k);
}

__device__ __forceinline__ void edge_sd(const int* __restrict__ ei, int Ereal,
                                        int e, int& s, int& d) {
  if (e < Ereal) { s = ei[e]; d = ei[Ereal + e]; }
  else           { s = e - Ereal; d = s; }        // appended self-loops
}

// ---------------------------------------------------------------------------
__global__ void fill_u32(unsigned* __restrict__ p, unsigned v, long n) {
  long i = (long)blockIdx.x * blockDim.x + threadIdx.x;
  long stride = (long)gridDim.x * blockDim.x;
  for (; i < n; i += stride) p[i] = v;
}

// ---- dense GEMM: Ho[N,128] = X[N,128] @ Wm[128,128] via V_WMMA_F32_16X16X4
// block = 256 thr = 8 waves; each wave owns one 16x16 tile; block covers a
// 16-row x 128-col strip. grid.x = N/16 (N = 40000 -> 2500 blocks).
__global__ __launch_bounds__(256) void gemm_wmma_f32(
    const float* __restrict__ X, const float* __restrict__ Wm,
    float* __restrict__ Ho) {
  const int wave = threadIdx.x >> 5;       // 0..7 -> column tile
  const int lane = threadIdx.x & 31;
  const int half = lane >> 4;              // 0 | 1
  const int l15  = lane & 15;
  const int row0 = blockIdx.x << 4;
  const int bcol = (wave << 4) + l15;
  const float* arow = X + (long)(row0 + l15) * Dch;

  v8f acc = {};
  for (int k = 0; k < Dch; k += 4) {
    const int ka = k + half * 2;           // A 16x4 f32 layout (ISA 7.12.2)
    v2f a, b;
    a.x = arow[ka];
    a.y = arow[ka + 1];
    b.x = Wm[(long)ka * Dch + bcol];       // B 4x16: rows striped like A halves
    b.y = Wm[(long)(ka + 1) * Dch + bcol];
    acc = __builtin_amdgcn_wmma_f32_16x16x4_f32(
        /*neg_a=*/false, a, /*neg_b=*/false, b,
        /*c_mod=*/(short)0, acc, /*reuse_a=*/false, /*reuse_b=*/false);
  }
  // C/D layout: VGPR v -> row = row0 + v + half*8, col = bcol
  float* orow = Ho + (long)(row0 + half * 8) * Dch + bcol;
#pragma unroll
  for (int v = 0; v < 8; ++v) orow[(long)v * Dch] = acc[v];
}

// ---- per-(node,head) attention logits -------------------------------------
__global__ void att_logits(const float* __restrict__ Hf,
                           const float* __restrict__ a_src,
                           const float* __restrict__ a_dst,
                           float* __restrict__ alsrc, float* __restrict__ aldst,
                           int Nn) {
  int i = blockIdx.x * blockDim.x + threadIdx.x;     // (node, head)
  if (i >= Nn * Hh) return;
  int n = i >> 2, h = i & 3;
  const float* hp = Hf + (long)n * Dch + h * Ch;
  const float* as = a_src + h * Ch;
  const float* ad = a_dst + h * Ch;
  float s0 = 0.f, s1 = 0.f;
#pragma unroll
  for (int c = 0; c < Ch; ++c) { float v = hp[c]; s0 += v * as[c]; s1 += v * ad[c]; }
  alsrc[i] = s0;
  aldst[i] = s1;
}

// ---- edge pass 1: segment max (as monotonic uint keys) --------------------
__global__ void edge_max(const int* __restrict__ ei, int Ereal, int Etot,
                         const float* __restrict__ alsrc,
                         const float* __restrict__ aldst,
                         unsigned* __restrict__ mkey) {
  int e = blockIdx.x * blockDim.x + threadIdx.x;
  if (e >= Etot) return;
  int s, d; edge_sd(ei, Ereal, e, s, d);
#pragma unroll
  for (int h = 0; h < Hh; ++h) {
    float sc = alsrc[s * Hh + h] + aldst[d * Hh + h];
    sc = sc > 0.f ? sc : sc * NEG_SLOPE;
    atomicMax(&mkey[d * Hh + h], fkey(sc));
  }
}

// ---- edge pass 2: sum of exp(score - max) ---------------------------------
__global__ void edge_expsum(const int* __restrict__ ei, int Ereal, int Etot,
                            const float* __restrict__ alsrc,
                            const float* __restrict__ aldst,
                            const unsigned* __restrict__ mkey,
                            float* __restrict__ ssum) {
  int e = blockIdx.x * blockDim.x + threadIdx.x;
  if (e >= Etot) return;
  int s, d; edge_sd(ei, Ereal, e, s, d);
#pragma unroll
  for (int h = 0; h < Hh; ++h) {
    float sc = alsrc[s * Hh + h] + aldst[d * Hh + h];
    sc = sc > 0.f ? sc : sc * NEG_SLOPE;
    float p = expf(sc - kinv(mkey[d * Hh + h]));
    atomicAdd(&ssum[d * Hh + h], p);
  }
}

// ---- edge pass 3: alpha-weighted scatter-add of source features -----------
// one 32-lane wave per edge; lane = channel-within-head; 4 heads per lane.
__global__ __launch_bounds__(256) void edge_aggregate(
    const int* __restrict__ ei, int Ereal, int Etot,
    const float* __restrict__ alsrc, const float* __restrict__ aldst,
    const unsigned* __restrict__ mkey, const float* __restrict__ ssum,
    const float* __restrict__ Hf, float* __restrict__ acc) {
  long gid = (long)blockIdx.x * blockDim.x + threadIdx.x;
  int e    = (int)(gid >> 5);
  int lane = (int)(gid & 31);
  if (e >= Etot) return;
  int s, d; edge_sd(ei, Ereal, e, s, d);
  const float* hsrc = Hf + (long)s * Dch;
  float* adst = acc + (long)d * Dch;
#pragma unroll
  for (int h = 0; h < Hh; ++h) {
    float sc = alsrc[s * Hh + h] + aldst[d * Hh + h];
    sc = sc > 0.f ? sc : sc * NEG_SLOPE;
    float p = expf(sc - kinv(mkey[d * Hh + h]));
    float alpha = p / (ssum[d * Hh + h] + 1e-16f);
    int ch = h * Ch + lane;
    atomicAdd(&adst[ch], alpha * hsrc[ch]);
  }
}

// ---- bias + BatchNorm(inference) + ELU ------------------------------------
__global__ void post_bn_elu(const float* __restrict__ acc,
                            const float* __restrict__ bias,
                            const float* __restrict__ gamma,
                            const float* __restrict__ beta,
                            const float* __restrict__ mean,
                            const float* __restrict__ var,
                            float* __restrict__ xo, long n) {
  long i = (long)blockIdx.x * blockDim.x + threadIdx.x;
  if (i >= n) return;
  int c = (int)(i & (Dch - 1));
  float v = acc[i] + bias[c];
  v = (v - mean[c]) * (gamma[c] / sqrtf(var[c] + BN_EPS)) + beta[c];
  xo[i] = v > 0.f ? v : expf(v) - 1.f;     // ELU(alpha=1)
}

// ---- global mean pool -----------------------------------------------------
__global__ void pool_accum(const float* __restrict__ x,
                           const int* __restrict__ batch,
                           float* __restrict__ sums, float* __restrict__ cnt,
                           long n) {
  long i = (long)blockIdx.x * blockDim.x + threadIdx.x;
  if (i >= n) return;
  int node = (int)(i >> 7);
  int c    = (int)(i & 127);
  int g    = batch[node];
  atomicAdd(&sums[(long)g * Dch + c], x[i]);
  if (c == 0) atomicAdd(&cnt[g], 1.0f);
}

__global__ void pool_final(const float* __restrict__ sums,
                           const float* __restrict__ cnt,
                           float* __restrict__ out, int G) {
  int i = blockIdx.x * blockDim.x + threadIdx.x;
  if (i >= G * Dch) return;
  int g = i >> 7;
  float c = cnt[g];
  out[i] = sums[i] / (c > 1.f ? c : 1.f);
}

// ---------------------------------------------------------------------------
extern "C" void kernel_launch(void* const* d_in, const int* in_sizes, int n_in,
                              void* d_out, int out_size, void* d_ws, size_t ws_size,
                              hipStream_t stream) {
  const float* x0    = (const float*)d_in[0];
  const int*   ei    = (const int*)d_in[1];
  const int*   batch = (const int*)d_in[2];
  const float* Ws    = (const float*)d_in[3];
  const float* a_src = (const float*)d_in[4];
  const float* a_dst = (const float*)d_in[5];
  const float* bias  = (const float*)d_in[6];
  const float* gamma = (const float*)d_in[7];
  const float* beta  = (const float*)d_in[8];
  const float* mean  = (const float*)d_in[9];
  const float* var   = (const float*)d_in[10];

  const int Nn    = in_sizes[0] / Dch;   // 40000
  const int Ereal = in_sizes[1] / 2;     // 640000
  const int Etot  = Ereal + Nn;          // + self-loops
  const int G     = out_size / Dch;      // 64

  char* ws = (char*)d_ws;
  float*    bufA  = (float*)ws;    ws += (size_t)Nn * Dch * 4;
  float*    bufB  = (float*)ws;    ws += (size_t)Nn * Dch * 4;
  float*    hbuf  = (float*)ws;    ws += (size_t)Nn * Dch * 4;
  unsigned* mkey  = (unsigned*)ws; ws += (size_t)Nn * Hh * 4;
  float*    ssum  = (float*)ws;    ws += (size_t)Nn * Hh * 4;
  float*    alsrc = (float*)ws;    ws += (size_t)Nn * Hh * 4;
  float*    aldst = (float*)ws;    ws += (size_t)Nn * Hh * 4;
  float*    psum  = (float*)ws;    ws += (size_t)G * Dch * 4;
  float*    pcnt  = (float*)ws;    ws += (size_t)G * 4;

  const float* xin  = x0;
  float*       xout = bufA;
  for (int l = 0; l < Lnum; ++l) {
    const float* Wl = Ws + (size_t)l * Dch * Dch;
    gemm_wmma_f32<<<Nn / 16, 256, 0, stream>>>(xin, Wl, hbuf);

    att_logits<<<(Nn * Hh + 255) / 256, 256, 0, stream>>>(
        hbuf, a_src + (size_t)l * Hh * Ch, a_dst + (size_t)l * Hh * Ch,
        alsrc, aldst, Nn);

    fill_u32<<<512, 256, 0, stream>>>(mkey, 0u, (long)Nn * Hh);          // key 0 == -inf
    fill_u32<<<512, 256, 0, stream>>>((unsigned*)ssum, 0u, (long)Nn * Hh);
    fill_u32<<<2048, 256, 0, stream>>>((unsigned*)xout, 0u, (long)Nn * Dch);

    edge_max<<<(Etot + 255) / 256, 256, 0, stream>>>(ei, Ereal, Etot, alsrc, aldst, mkey);
    edge_expsum<<<(Etot + 255) / 256, 256, 0, stream>>>(ei, Ereal, Etot, alsrc, aldst, mkey, ssum);
    {
      long t = (long)Etot * 32;
      edge_aggregate<<<(int)((t + 255) / 256), 256, 0, stream>>>(
          ei, Ereal, Etot, alsrc, aldst, mkey, ssum, hbuf, xout);
    }

    post_bn_elu<<<(Nn * Dch + 255) / 256, 256, 0, stream>>>(
        xout, bias + (size_t)l * Dch, gamma + (size_t)l * Dch,
        beta + (size_t)l * Dch, mean + (size_t)l * Dch, var + (size_t)l * Dch,
        xout, (long)Nn * Dch);

    xin  = xout;
    xout = (xout == bufA) ? bufB : bufA;
  }

  fill_u32<<<32, 256, 0, stream>>>((unsigned*)psum, 0u, (long)G * Dch);
  fill_u32<<<1, 64, 0, stream>>>((unsigned*)pcnt, 0u, G);
  pool_accum<<<(Nn * Dch + 255) / 256, 256, 0, stream>>>(xin, batch, psum, pcnt, (long)Nn * Dch);
  pool_final<<<(G * Dch + 255) / 256, 256, 0, stream>>>(psum, pcnt, (float*)d_out, G);
}